// Net_63273458205054
// MI455X (gfx1250) — compile-verified
//
#include <hip/hip_runtime.h>
#include <hip/hip_bf16.h>
#include <math.h>

typedef float v2f __attribute__((ext_vector_type(2)));
typedef float v8f __attribute__((ext_vector_type(8)));

#define EPS 1e-5f

// ---------------------------------------------------------------------------
// AdderNet L1 pseudo-conv: out[b,o,h,w] = -sum_{c,kh,kw} |x[b,c,h+kh-1,w+kw-1] - w[o, c*9+kh*3+kw]|
// SAME padding (zeros) -> padded taps contribute |w|.
// ---------------------------------------------------------------------------
__global__ __launch_bounds__(256) void adder2d_kernel(
    const float* __restrict__ x, const float* __restrict__ wt, float* __restrict__ out,
    int B, int C, int H, int W, int Co)
{
    int idx = blockIdx.x * 256 + threadIdx.x;
    int total = B * Co * H * W;
    if (idx >= total) return;
    int ww = idx % W;
    int t  = idx / W;
    int h  = t % H;  t /= H;
    int o  = t % Co;
    int b  = t / Co;

    const float* wrow = wt + o * C * 9;
    float s = 0.0f;
    for (int c = 0; c < C; ++c) {
        const float* xc = x + ((size_t)(b * C + c)) * H * W;
        const float* wc = wrow + c * 9;
        #pragma unroll
        for (int kh = 0; kh < 3; ++kh) {
            int ih = h + kh - 1;
            bool vy = (ih >= 0) && (ih < H);
            #pragma unroll
            for (int kw = 0; kw < 3; ++kw) {
                int iw = ww + kw - 1;
                float xv = (vy && iw >= 0 && iw < W) ? xc[ih * W + iw] : 0.0f;
                s += fabsf(xv - wc[kh * 3 + kw]);
            }
        }
    }
    out[idx] = -s;
}

// ---------------------------------------------------------------------------
// Per-channel batch stats (mean, rsqrt(var+eps)); biased variance over (N,H,W).
// One block per channel; LDS tree reduction.
// ---------------------------------------------------------------------------
__global__ __launch_bounds__(256) void bn_stats_kernel(
    const float* __restrict__ a, float* __restrict__ mu, float* __restrict__ rstd,
    int B, int C, int HW)
{
    __shared__ float ssum[256];
    __shared__ float ssq[256];
    int c = blockIdx.x;
    int n = B * HW;
    float s = 0.0f, q = 0.0f;
    for (int i = threadIdx.x; i < n; i += 256) {
        int b = i / HW;
        int r = i - b * HW;
        float v = a[((size_t)(b * C + c)) * HW + r];
        s += v;
        q += v * v;
    }
    ssum[threadIdx.x] = s;
    ssq[threadIdx.x]  = q;
    __syncthreads();
    for (int st = 128; st > 0; st >>= 1) {
        if (threadIdx.x < (unsigned)st) {
            ssum[threadIdx.x] += ssum[threadIdx.x + st];
            ssq[threadIdx.x]  += ssq[threadIdx.x + st];
        }
        __syncthreads();
    }
    if (threadIdx.x == 0) {
        float inv_n = 1.0f / (float)n;
        float m   = ssum[0] * inv_n;
        float var = ssq[0] * inv_n - m * m;
        mu[c]   = m;
        rstd[c] = rsqrtf(var + EPS);
    }
}

// ---------------------------------------------------------------------------
// Fused BN(train) + ReLU + 2x2 maxpool.  out: [B,C,H/2,W/2]
// ---------------------------------------------------------------------------
__global__ __launch_bounds__(256) void bn_relu_pool_kernel(
    const float* __restrict__ a, const float* __restrict__ mu, const float* __restrict__ rstd,
    const float* __restrict__ gamma, const float* __restrict__ beta,
    float* __restrict__ out, int B, int C, int H, int W)
{
    int Ho = H >> 1, Wo = W >> 1;
    int idx = blockIdx.x * 256 + threadIdx.x;
    int total = B * C * Ho * Wo;
    if (idx >= total) return;
    int wo = idx % Wo;
    int t  = idx / Wo;
    int ho = t % Ho;  t /= Ho;
    int c  = t % C;
    int b  = t / C;

    float m = mu[c], rs = rstd[c], g = gamma[c], bt = beta[c];
    float scale = rs * g;
    float shift = bt - m * scale;

    const float* base = a + (((size_t)(b * C + c)) * H + ho * 2) * W + wo * 2;
    float v0 = fmaxf(base[0]     * scale + shift, 0.0f);
    float v1 = fmaxf(base[1]     * scale + shift, 0.0f);
    float v2 = fmaxf(base[W]     * scale + shift, 0.0f);
    float v3 = fmaxf(base[W + 1] * scale + shift, 0.0f);
    out[idx] = fmaxf(fmaxf(v0, v1), fmaxf(v2, v3));
}

// ---------------------------------------------------------------------------
// Final contraction: out[b,o] = sum_k h[b,k] * w4[o,k] + bias[o]
// M=128, N=10 (padded to 16), K=1024, full-precision V_WMMA_F32_16X16X4_F32.
// One wave (32 threads) per 16-row tile of M; 8 blocks total.
//
// fp32 WMMA operand layout (ISA 7.12.2):
//  A (16x4):  lane<16 -> row=lane, K={k0,k0+1}; lane>=16 -> row=lane-16, K={k0+2,k0+3}
//  B (4x16):  lane<16 -> col=lane, K={k0,k0+1}; lane>=16 -> col=lane-16, K={k0+2,k0+3}
//  D (16x16): VGPR r, lane<16 -> M=r, N=lane; lane>=16 -> M=r+8, N=lane-16
// Columns 10..15 of B are clamped to a valid row; their D columns are never
// stored (D column n depends only on B column n), so no masking is needed and
// EXEC stays all-1s through every WMMA.
// ---------------------------------------------------------------------------
__global__ __launch_bounds__(32) void gemm_wmma_kernel(
    const float* __restrict__ Hm, const float* __restrict__ W4,
    const float* __restrict__ bias, float* __restrict__ out)
{
    const int K = 1024;
    int lane = threadIdx.x;
    int half = lane >> 4;       // 0 or 1
    int lid  = lane & 15;
    int mrow = blockIdx.x * 16 + lid;
    int bcol = (lid < 10) ? lid : 9;   // clamp, result cols >=10 discarded
    int koff = half * 2;

    const float* arow = Hm + (size_t)mrow * K + koff;
    const float* brow = W4 + (size_t)bcol * K + koff;

    v8f acc = {};
    for (int k0 = 0; k0 < K; k0 += 4) {
        v2f av = *(const v2f*)(arow + k0);
        v2f bv = *(const v2f*)(brow + k0);
        acc = __builtin_amdgcn_wmma_f32_16x16x4_f32(
            false, av, false, bv, (short)0, acc, false, false);
    }

    if (lid < 10) {
        float bo = bias[lid];
        #pragma unroll
        for (int r = 0; r < 8; ++r) {
            int m = blockIdx.x * 16 + r + half * 8;
            out[m * 10 + lid] = acc[r] + bo;
        }
    }
}

// ---------------------------------------------------------------------------
// Workspace layout (floats):
//   a1 : [128,16,32,32] @ 0         (2097152)
//   p1 : [128,16,16,16] @ 2097152   (524288)
//   a2 : [128,32,16,16] @ 2621440   (1048576)
//   p2 : [128,32, 8, 8] @ 3670016   (262144)
//   a3 : [128,64, 8, 8] @ 3932160   (524288)
//   p3 : [128,64, 4, 4] @ 4456448   (131072)
//   mu : [64]           @ 4587520
//   rstd:[64]           @ 4587584
// Total 4587648 floats = ~18.4 MB
// ---------------------------------------------------------------------------
extern "C" void kernel_launch(void* const* d_in, const int* in_sizes, int n_in,
                              void* d_out, int out_size, void* d_ws, size_t ws_size,
                              hipStream_t stream)
{
    (void)in_sizes; (void)n_in; (void)out_size; (void)ws_size;

    const float* x     = (const float*)d_in[0];
    const float* w1    = (const float*)d_in[1];
    const float* g1    = (const float*)d_in[2];
    const float* b1    = (const float*)d_in[3];
    const float* w2    = (const float*)d_in[4];
    const float* g2    = (const float*)d_in[5];
    const float* b2    = (const float*)d_in[6];
    const float* w3    = (const float*)d_in[7];
    const float* g3    = (const float*)d_in[8];
    const float* b3    = (const float*)d_in[9];
    const float* w4    = (const float*)d_in[10];
    const float* bias4 = (const float*)d_in[11];
    float* out = (float*)d_out;

    float* ws  = (float*)d_ws;
    float* a1  = ws;
    float* p1  = ws + 2097152;
    float* a2  = ws + 2621440;
    float* p2  = ws + 3670016;
    float* a3  = ws + 3932160;
    float* p3  = ws + 4456448;
    float* mu  = ws + 4587520;
    float* rsd = ws + 4587584;

    const int B = 128;

    // ---- Layer 1: [128,3,32,32] -> [128,16,16,16]
    {
        int C = 3, H = 32, W = 32, Co = 16;
        int tot = B * Co * H * W;
        adder2d_kernel<<<(tot + 255) / 256, 256, 0, stream>>>(x, w1, a1, B, C, H, W, Co);
        bn_stats_kernel<<<Co, 256, 0, stream>>>(a1, mu, rsd, B, Co, H * W);
        int tpo = B * Co * (H / 2) * (W / 2);
        bn_relu_pool_kernel<<<(tpo + 255) / 256, 256, 0, stream>>>(a1, mu, rsd, g1, b1, p1, B, Co, H, W);
    }

    // ---- Layer 2: [128,16,16,16] -> [128,32,8,8]
    {
        int C = 16, H = 16, W = 16, Co = 32;
        int tot = B * Co * H * W;
        adder2d_kernel<<<(tot + 255) / 256, 256, 0, stream>>>(p1, w2, a2, B, C, H, W, Co);
        bn_stats_kernel<<<Co, 256, 0, stream>>>(a2, mu, rsd, B, Co, H * W);
        int tpo = B * Co * (H / 2) * (W / 2);
        bn_relu_pool_kernel<<<(tpo + 255) / 256, 256, 0, stream>>>(a2, mu, rsd, g2, b2, p2, B, Co, H, W);
    }

    // ---- Layer 3: [128,32,8,8] -> [128,64,4,4]
    {
        int C = 32, H = 8, W = 8, Co = 64;
        int tot = B * Co * H * W;
        adder2d_kernel<<<(tot + 255) / 256, 256, 0, stream>>>(p2, w3, a3, B, C, H, W, Co);
        bn_stats_kernel<<<Co, 256, 0, stream>>>(a3, mu, rsd, B, Co, H * W);
        int tpo = B * Co * (H / 2) * (W / 2);
        bn_relu_pool_kernel<<<(tpo + 255) / 256, 256, 0, stream>>>(a3, mu, rsd, g3, b3, p3, B, Co, H, W);
    }

    // ---- Final dense contraction via fp32 WMMA: [128,1024] x [10,1024]^T + bias
    gemm_wmma_kernel<<<8, 32, 0, stream>>>(p3, w4, bias4, out);
}